// KNNModel_60370060313142
// MI455X (gfx1250) — compile-verified
//
#include <hip/hip_runtime.h>
#include <cstdint>
#include <cstddef>

#define AS1 __attribute__((address_space(1)))
#define AS3 __attribute__((address_space(3)))

#if __has_builtin(__builtin_amdgcn_global_load_async_to_lds_b128) && \
    __has_builtin(__builtin_amdgcn_s_wait_asynccnt)
#define USE_ASYNC_LDS 1
#else
#define USE_ASYNC_LDS 0
#endif

typedef __attribute__((ext_vector_type(4))) float f32x4;
typedef __attribute__((ext_vector_type(4))) int   i32x4;

static constexpr int   KNN       = 32;
static constexpr int   BLOCK     = 256;
static constexpr float SIM_T     = 0.7f;
static constexpr float VIRAL_T   = 0.2f;

__global__ __launch_bounds__(BLOCK) void knn_viral_pred_kernel(
    const float*         __restrict__ sims,        // (B, 32) f32
    const int*           __restrict__ knns,        // (B, 32) i32
    const unsigned char* __restrict__ if_viral,    // (N,) bool (1 byte)
    const float*         __restrict__ retweet_cnt, // (N,) f32
    float*               __restrict__ out,         // (B,) f32
    int B)
{
    const int row = (int)(blockIdx.x * BLOCK + threadIdx.x);
    const int r   = row < B ? row : B - 1;          // clamp; keeps waves uniform
    const size_t rb = (size_t)r * KNN;

    float s[KNN];
    int   j[KNN];

#if USE_ASYNC_LDS
    // Stage the two streaming rows via the CDNA5 async global->LDS path:
    // 8x GLOBAL_LOAD_ASYNC_TO_LDS_B128 per thread, tracked by ASYNCcnt.
    __shared__ __align__(16) float lds_s[BLOCK * KNN];
    __shared__ __align__(16) int   lds_j[BLOCK * KNN];
    {
        AS1 i32x4* gs = (AS1 i32x4*)(sims + rb);
        AS1 i32x4* gj = (AS1 i32x4*)(knns + rb);
        AS3 i32x4* ls = (AS3 i32x4*)&lds_s[threadIdx.x * KNN];
        AS3 i32x4* lj = (AS3 i32x4*)&lds_j[threadIdx.x * KNN];
        // INST_OFFSET applies to both global and LDS addresses (ISA 08 §4.4).
        __builtin_amdgcn_global_load_async_to_lds_b128(gs, ls,  0, 0);
        __builtin_amdgcn_global_load_async_to_lds_b128(gs, ls, 16, 0);
        __builtin_amdgcn_global_load_async_to_lds_b128(gs, ls, 32, 0);
        __builtin_amdgcn_global_load_async_to_lds_b128(gs, ls, 48, 0);
        __builtin_amdgcn_global_load_async_to_lds_b128(gj, lj,  0, 0);
        __builtin_amdgcn_global_load_async_to_lds_b128(gj, lj, 16, 0);
        __builtin_amdgcn_global_load_async_to_lds_b128(gj, lj, 32, 0);
        __builtin_amdgcn_global_load_async_to_lds_b128(gj, lj, 48, 0);
        __builtin_amdgcn_s_wait_asynccnt(0);
        asm volatile("" ::: "memory");   // keep LDS reads below the wait
    }
    #pragma unroll
    for (int q = 0; q < KNN / 4; ++q) {
        f32x4 fs = ((const f32x4*)&lds_s[threadIdx.x * KNN])[q];   // ds_load_b128
        i32x4 js = ((const i32x4*)&lds_j[threadIdx.x * KNN])[q];
        #pragma unroll
        for (int c4 = 0; c4 < 4; ++c4) { s[4*q + c4] = fs[c4]; j[4*q + c4] = js[c4]; }
    }
#else
    // Fallback: direct non-temporal b128 streaming loads (data is read once;
    // NT keeps the 10 MB gather tables resident in L2).
    #pragma unroll
    for (int q = 0; q < KNN / 4; ++q) {
        f32x4 fs = __builtin_nontemporal_load(((const f32x4*)(sims + rb)) + q);
        i32x4 js = __builtin_nontemporal_load(((const i32x4*)(knns + rb)) + q);
        #pragma unroll
        for (int c4 = 0; c4 < 4; ++c4) { s[4*q + c4] = fs[c4]; j[4*q + c4] = js[c4]; }
    }
#endif

    // Pass 1: masks + gathers (L2-resident tables).
    unsigned keepm = 0u, viralm = 0u;
    float c[KNN];
    #pragma unroll
    for (int k = 0; k < KNN; ++k) {
        const int idx = j[k];
        const bool keep = s[k] > SIM_T;
        const bool vir  = keep && (if_viral[idx] != 0);
        keepm  |= (unsigned)keep << k;
        viralm |= (unsigned)vir  << k;
        c[k] = retweet_cnt[idx];                    // global_load_b32 gather
    }

    // Pass 2: masked softmax-weighted sum (registers only).
    float pred = 0.0f;
    if (viralm != 0u) {
        float m = -1e30f;
        #pragma unroll
        for (int k = 0; k < KNN; ++k) {
            const float cand = ((viralm >> k) & 1u) ? s[k] : -1e30f;
            m = fmaxf(m, cand);
        }
        float se = 0.0f, sec = 0.0f;
        #pragma unroll
        for (int k = 0; k < KNN; ++k) {
            const float e = ((viralm >> k) & 1u) ? __expf(s[k] - m) : 0.0f;
            se  += e;
            sec += e * c[k];
        }
        const int nk = __popc(keepm);
        const int nv = __popc(viralm);
        const float ratio = (float)nv / (float)nk;  // nk >= nv >= 1 here
        if (ratio >= VIRAL_T)
            pred = sec / fmaxf(se, 1e-30f);
    }

    if (row < B) out[row] = pred;
}

extern "C" void kernel_launch(void* const* d_in, const int* in_sizes, int n_in,
                              void* d_out, int out_size, void* d_ws, size_t ws_size,
                              hipStream_t stream) {
    const float*         sims        = (const float*)d_in[0];
    const int*           knns        = (const int*)d_in[1];
    const unsigned char* if_viral    = (const unsigned char*)d_in[2];
    const float*         retweet_cnt = (const float*)d_in[3];
    float*               out         = (float*)d_out;

    const int B = in_sizes[0] / KNN;                // sims is (B, 32)
    const int grid = (B + BLOCK - 1) / BLOCK;       // 2048 blocks of 256
    knn_viral_pred_kernel<<<grid, BLOCK, 0, stream>>>(
        sims, knns, if_viral, retweet_cnt, out, B);
    (void)d_ws; (void)ws_size; (void)n_in; (void)out_size;
}